// HybridSelfAttention_65481071400492
// MI455X (gfx1250) — compile-verified
//
#include <hip/hip_runtime.h>
#include <hip/hip_bf16.h>

// ---------------------------------------------------------------------------
// HybridSelfAttention (RBF-kernel attention), MI455X / gfx1250, wave32.
//   logits[t,j,b] = 2*gamma * (q_t . k_j) - gamma*||k_j||^2   (||q||^2 cancels
//   in softmax), out = softmax_j(logits) @ V.
// All GEMM work goes through V_WMMA_F32_16X16X4_F32 (fp32 matrix pipe) to stay
// numerically faithful to the fp32 reference.  All hot-loop memory accesses
// are forced into the GLOBAL address space so they lower to global_load/store
// (not flat_*, which would also consume DScnt / the LDS datapath).
// ---------------------------------------------------------------------------

#define S_LEN 2048
#define B_SZ  8
#define E_DIM 1024
constexpr float GAMMA = 0.5f;

typedef __attribute__((ext_vector_type(2))) float v2f;
typedef __attribute__((ext_vector_type(8))) float v8f;

#define GAS __attribute__((address_space(1)))

static __device__ __forceinline__ v2f load_g2(const float* p) {
  return *(const GAS v2f*)(p);
}
static __device__ __forceinline__ float load_g1(const float* p) {
  return *(const GAS float*)(p);
}
static __device__ __forceinline__ void store_g1(float* p, float x) {
  *(GAS float*)(p) = x;
}

// D = A(16x4) * B(4x16) + C(16x16), fp32, wave32.
// Layouts (ISA 7.12.2):
//   A: lane(l) -> M = l%16 ; VGPR g -> K = 2*(l/16)+g        (v2f per lane)
//   B: lane(l) -> N = l%16 ; VGPR g -> K = 2*(l/16)+g        (v2f per lane)
//   C: lane(l) -> N = l%16 ; VGPR g -> M = g + 8*(l/16)      (v8f per lane)
static __device__ __forceinline__ v8f wmma_f32(v2f a, v2f b, v8f c) {
  return __builtin_amdgcn_wmma_f32_16x16x4_f32(false, a, false, b,
                                               (short)0, c, false, false);
}

// ---------------------------------------------------------------------------
// Kernel 1: fused QKV projection.  q/k/v[r,e] = sum_c x[r,c]*W[e,c] + b[e]
// with r = s*B + b over the flattened (S,B) rows.  One wave per 16x64 output
// tile; the x-tile A-fragment is reused across {Wq,Wk,Wv} x 4 N-subtiles.
// Base pointers are hoisted and the K loop is unrolled 4x so the inner loads
// use constant offset: immediates instead of per-load 64-bit address math.
// ---------------------------------------------------------------------------
__global__ __launch_bounds__(32)
void qkv_proj_kernel(const float* __restrict__ x,
                     const float* __restrict__ Wq, const float* __restrict__ bq,
                     const float* __restrict__ Wk, const float* __restrict__ bk,
                     const float* __restrict__ Wv, const float* __restrict__ bv,
                     float* __restrict__ q, float* __restrict__ k,
                     float* __restrict__ v) {
  const int lane = threadIdx.x;
  const int half = lane >> 4;      // 0: lanes 0-15, 1: lanes 16-31
  const int lm   = lane & 15;
  const int m0   = blockIdx.x * 16;   // row-tile base (over S*B)
  const int n0   = blockIdx.y * 64;   // col-tile base (over E)

  v8f aq[4], ak[4], av[4];
  const float* wqp[4];
  const float* wkp[4];
  const float* wvp[4];
  #pragma unroll
  for (int ns = 0; ns < 4; ++ns) {
    const int col = n0 + ns * 16 + lm;  // C layout: col = lane%16
    const float biq = load_g1(bq + col);
    const float bik = load_g1(bk + col);
    const float biv = load_g1(bv + col);
    #pragma unroll
    for (int g = 0; g < 8; ++g) { aq[ns][g] = biq; ak[ns][g] = bik; av[ns][g] = biv; }
    const size_t wo = (size_t)col * E_DIM + 2 * half;  // W[e,c] row-major
    wqp[ns] = Wq + wo;
    wkp[ns] = Wk + wo;
    wvp[ns] = Wv + wo;
  }

  const float* xp = x + (size_t)(m0 + lm) * E_DIM + 2 * half;
  for (int kk = 0; kk < E_DIM; kk += 16) {
    #pragma unroll
    for (int u = 0; u < 4; ++u) {                     // 4 k-steps of K=4
      const v2f a = load_g2(xp + 4 * u);              // A frag: rows of x
      #pragma unroll
      for (int ns = 0; ns < 4; ++ns) {
        aq[ns] = wmma_f32(a, load_g2(wqp[ns] + 4 * u), aq[ns]);
        ak[ns] = wmma_f32(a, load_g2(wkp[ns] + 4 * u), ak[ns]);
        av[ns] = wmma_f32(a, load_g2(wvp[ns] + 4 * u), av[ns]);
      }
    }
    xp += 16;
    #pragma unroll
    for (int ns = 0; ns < 4; ++ns) { wqp[ns] += 16; wkp[ns] += 16; wvp[ns] += 16; }
  }

  #pragma unroll
  for (int ns = 0; ns < 4; ++ns) {
    const int col = n0 + ns * 16 + lm;
    #pragma unroll
    for (int g = 0; g < 8; ++g) {
      const size_t row = (size_t)(m0 + g + 8 * half);
      store_g1(q + row * E_DIM + col, aq[ns][g]);
      store_g1(k + row * E_DIM + col, ak[ns][g]);
      store_g1(v + row * E_DIM + col, av[ns][g]);
    }
  }
}

// ---------------------------------------------------------------------------
// Kernel 2: gk2[r] = GAMMA * ||k[r,:]||^2   (one WG per row, LDS tree reduce)
// ---------------------------------------------------------------------------
__global__ __launch_bounds__(256)
void rownorm_kernel(const float* __restrict__ k, float* __restrict__ gk2) {
  __shared__ float red[256];
  const float* kr = k + (size_t)blockIdx.x * E_DIM;
  float s = 0.f;
  for (int e = threadIdx.x; e < E_DIM; e += 256) {
    const float t = load_g1(kr + e);
    s += t * t;
  }
  red[threadIdx.x] = s;
  __syncthreads();
  for (int off = 128; off > 0; off >>= 1) {
    if (threadIdx.x < off) red[threadIdx.x] += red[threadIdx.x + off];
    __syncthreads();
  }
  if (threadIdx.x == 0) store_g1(gk2 + blockIdx.x, GAMMA * red[0]);
}

// ---------------------------------------------------------------------------
// Kernel 3: flash attention.  WG = 8 waves, one 16-query tile of one batch.
// Wave w owns E-slice [w*128, w*128+128): partial scores over its slice
// (cross-wave reduced in LDS), then P·V into private 16x128 accumulators.
// Softmax: 16 threads do row-max / row-sum; the 256 expf's run in parallel
// (one per thread, logits kept in registers).
// ---------------------------------------------------------------------------
__global__ __launch_bounds__(256)
void attn_kernel(const float* __restrict__ q, const float* __restrict__ k,
                 const float* __restrict__ v, const float* __restrict__ gk2,
                 float* __restrict__ out) {
  const int tid  = threadIdx.x;
  const int w    = tid >> 5;
  const int lane = tid & 31;
  const int half = lane >> 4;
  const int lm   = lane & 15;
  const int b    = blockIdx.y;
  const int t0   = blockIdx.x * 16;
  const int e0   = w * 128;                 // this wave's E-slice base

  __shared__ float sS[8][256];              // per-wave 16x16 score partials
  __shared__ float sP[256];                 // logits -> probabilities (16x16)
  __shared__ float sScale[16];              // per-row exp(m_old - m_new)
  __shared__ float sM[16];                  // running row max
  __shared__ float sL[16];                  // running row sum

  if (tid < 16) { sM[tid] = -__builtin_inff(); sL[tid] = 0.f; }
  __syncthreads();

  // Preload Q slice (16 x 128) as 32 A-fragments, pre-scaled by 2*gamma.
  v2f qf[32];
  {
    const float* qrow =
        q + ((size_t)(t0 + lm) * B_SZ + b) * E_DIM + e0 + 2 * half;
    #pragma unroll
    for (int i = 0; i < 32; ++i) qf[i] = load_g2(qrow + 4 * i) * (2.0f * GAMMA);
  }

  v8f acc[8];                               // out slice 16 x 128
  #pragma unroll
  for (int ns = 0; ns < 8; ++ns) acc[ns] = (v8f){};

  for (int j0 = 0; j0 < S_LEN; j0 += 16) {
    // --- partial scores over this wave's E slice: S += Q_slice * K_slice^T
    v8f sc = (v8f){};
    const float* krow =
        k + ((size_t)(j0 + lm) * B_SZ + b) * E_DIM + e0 + 2 * half;  // B: N=j
    #pragma unroll
    for (int i = 0; i < 32; ++i)
      sc = wmma_f32(qf[i], load_g2(krow + 4 * i), sc);

    #pragma unroll
    for (int g = 0; g < 8; ++g)
      sS[w][(g + 8 * half) * 16 + lm] = sc[g];
    __syncthreads();                               // (1) partials ready

    // --- cross-wave reduce + key bias (one element per thread)
    float logit;
    {
      float s = 0.f;
      #pragma unroll
      for (int ww = 0; ww < 8; ++ww) s += sS[ww][tid];
      logit = s - load_g1(gk2 + (size_t)(j0 + (tid & 15)) * B_SZ + b);
      sP[tid] = logit;
    }
    __syncthreads();                               // (2) logits ready

    // --- row max + rescale factor (16 threads, one per query row)
    float scl_row = 0.f;                           // valid only for tid<16
    if (tid < 16) {
      const int row = tid;
      float mrow = -__builtin_inff();
      #pragma unroll
      for (int c = 0; c < 16; ++c) mrow = fmaxf(mrow, sP[row * 16 + c]);
      const float mnew = fmaxf(sM[row], mrow);
      scl_row = __expf(sM[row] - mnew);            // 0 on first tile
      sM[row] = mnew;
      sScale[row] = scl_row;
    }
    __syncthreads();                               // (3) sM/sScale ready

    // --- parallel exp: each thread converts its own logit
    sP[tid] = __expf(logit - sM[tid >> 4]);
    __syncthreads();                               // (4) probabilities ready

    // --- row sum update (overlaps with other waves' rescale/PV below)
    if (tid < 16) {
      float psum = 0.f;
      #pragma unroll
      for (int c = 0; c < 16; ++c) psum += sP[tid * 16 + c];
      sL[tid] = sL[tid] * scl_row + psum;
    }

    // --- rescale accumulators by per-row factor
    float f[8];
    #pragma unroll
    for (int g = 0; g < 8; ++g) f[g] = sScale[g + 8 * half];
    #pragma unroll
    for (int ns = 0; ns < 8; ++ns)
      #pragma unroll
      for (int g = 0; g < 8; ++g) acc[ns][g] *= f[g];

    // --- acc += P(16x16) * V_tile(16 x 128)
    #pragma unroll
    for (int c = 0; c < 4; ++c) {
      const v2f pf = *(const v2f*)&sP[lm * 16 + 4 * c + 2 * half];  // A frag
      const int jA = j0 + 4 * c + 2 * half;                          // B: K=j
      const float* vbase = v + ((size_t)jA * B_SZ + b) * E_DIM + e0 + lm;
      #pragma unroll
      for (int ns = 0; ns < 8; ++ns) {
        v2f vf;
        vf.x = load_g1(vbase + ns * 16);
        vf.y = load_g1(vbase + (size_t)B_SZ * E_DIM + ns * 16); // next key row
        acc[ns] = wmma_f32(pf, vf, acc[ns]);
      }
    }
    // No trailing barrier: next-iter LDS writes all sit behind barriers that
    // require every wave to have finished this iteration's reads.
  }
  __syncthreads();  // make the final sL updates visible to all waves

  // --- normalize and store
  float linv[8];
  #pragma unroll
  for (int g = 0; g < 8; ++g) linv[g] = 1.0f / sL[g + 8 * half];
  #pragma unroll
  for (int ns = 0; ns < 8; ++ns) {
    const int e = e0 + ns * 16 + lm;
    #pragma unroll
    for (int g = 0; g < 8; ++g) {
      const int trow = t0 + g + 8 * half;
      store_g1(out + ((size_t)trow * B_SZ + b) * E_DIM + e, acc[ns][g] * linv[g]);
    }
  }
}

// ---------------------------------------------------------------------------
extern "C" void kernel_launch(void* const* d_in, const int* in_sizes, int n_in,
                              void* d_out, int out_size, void* d_ws, size_t ws_size,
                              hipStream_t stream) {
  (void)in_sizes; (void)n_in; (void)out_size; (void)ws_size;
  const float* x  = (const float*)d_in[0];
  const float* Wq = (const float*)d_in[1];
  const float* bq = (const float*)d_in[2];
  const float* Wk = (const float*)d_in[3];
  const float* bk = (const float*)d_in[4];
  const float* Wv = (const float*)d_in[5];
  const float* bv = (const float*)d_in[6];
  float* out = (float*)d_out;

  const size_t mat = (size_t)S_LEN * B_SZ * E_DIM;
  float* q   = (float*)d_ws;
  float* k   = q + mat;
  float* v   = k + mat;
  float* gk2 = v + mat;      // S*B floats of gamma*||k||^2

  dim3 g1((S_LEN * B_SZ) / 16, E_DIM / 64);
  qkv_proj_kernel<<<g1, 32, 0, stream>>>(x, Wq, bq, Wk, bk, Wv, bv, q, k, v);

  rownorm_kernel<<<S_LEN * B_SZ, 256, 0, stream>>>(k, gk2);

  dim3 g3(S_LEN / 16, B_SZ);
  attn_kernel<<<g3, 256, 0, stream>>>(q, k, v, gk2, out);
}